// CSILoss_38749194945153
// MI455X (gfx1250) — compile-verified
//
#include <hip/hip_runtime.h>
#include <hip/hip_bf16.h>
#include <math.h>

// CSI loss for MI455X (gfx1250, wave32).
// Phase 1: one 256-thread block per antenna-pair row (1024 subcarriers).
//   - single coalesced read of all 4 inputs (268 MB total = HBM roofline floor)
//   - raw magnitudes staged in LDS so the JS-divergence second pass never
//     touches HBM again
//   - all 256->1 row reductions done with V_WMMA_F32_16X16X4_F32 (B = ones
//     => D rows are row-sums of A), offloading reduction adds to the matrix
//     pipe which co-executes with the transcendental-heavy VALU stream
// Phase 2: one block deterministically folds the 16384 per-row losses.

typedef __attribute__((ext_vector_type(2))) float v2f;
typedef __attribute__((ext_vector_type(8))) float v8f;

#define EPSF 1e-12f
#define MAX_MAGF 100.0f
#define SUBC 1024
#define BLK 256
#define PER_THREAD 4  // SUBC / BLK

// Sum 256 floats living in LDS using the matrix pipe. Must be executed by a
// full wave (lanes 0..31 of wave 0) so EXEC is all ones for the WMMA.
// A is 16x4 f32: lanes 0-15 hold K={0,1} in the two A VGPRs, lanes 16-31 hold
// K={2,3}; row M = lane & 15. Each A element pre-folds 4 of the 256 partials,
// so one WMMA + 8 lane-adds + 1 shuffle completes the reduction.
__device__ __forceinline__ float wmma_reduce256(const float* part) {
  const int lane = threadIdx.x & 31;
  const int m = lane & 15;
  const int kb = (lane < 16) ? 0 : 2;
  float a0 = 0.0f, a1 = 0.0f;
#pragma unroll
  for (int w = 0; w < 4; ++w) {
    a0 += part[m * 4 + kb + 0 + 64 * w];
    a1 += part[m * 4 + kb + 1 + 64 * w];
  }
  v2f A;
  A.x = a0;
  A.y = a1;
  v2f B;
  B.x = 1.0f;
  B.y = 1.0f;
  v8f C = {};
  // 8 args: (neg_a, A, neg_b, B, c_mod, C, reuse_a, reuse_b)
  v8f D = __builtin_amdgcn_wmma_f32_16x16x4_f32(false, A, false, B, (short)0,
                                                C, false, false);
  // Columns of D are identical (B == ones); sum this half's 8 M-rows, then
  // add the other lane-half (M 8..15 vs 0..7).
  float s = D[0] + D[1] + D[2] + D[3] + D[4] + D[5] + D[6] + D[7];
  s += __shfl_xor(s, 16, 32);
  return s;
}

__global__ __launch_bounds__(BLK) void csi_loss_rows(
    const float* __restrict__ g_pr, const float* __restrict__ g_pi,
    const float* __restrict__ g_tr, const float* __restrict__ g_ti,
    float* __restrict__ row_loss) {
  __shared__ float s_magP[SUBC];      // raw |pred| per subcarrier
  __shared__ float s_magT[SUBC];      // raw |target|
  __shared__ float s_part[11][BLK];   // per-thread partial sums
  __shared__ float s_PQ[2];

  const int tid = threadIdx.x;
  const size_t base = (size_t)blockIdx.x * SUBC;

  float d2 = 0.f, spm = 0.f, stm = 0.f, spm2 = 0.f, stm2 = 0.f;
  float sph = 0.f, scr = 0.f, sP = 0.f, sQ = 0.f;

#pragma unroll
  for (int w = 0; w < PER_THREAD; ++w) {
    const int s = tid + BLK * w;
    const float pr = g_pr[base + s];
    const float pi = g_pi[base + s];
    const float tr = g_tr[base + s];
    const float ti = g_ti[base + s];

    const float pmag = sqrtf(fmaf(pr, pr, pi * pi));
    const float tmag = sqrtf(fmaf(tr, tr, ti * ti));
    s_magP[s] = pmag;
    s_magT[s] = tmag;
    sP += pmag + EPSF;  // row sums for JS normalization (unclipped mags)
    sQ += tmag + EPSF;

    const float pm = fminf(fmaxf(pmag, EPSF), MAX_MAGF);
    const float tm = fminf(fmaxf(tmag, EPSF), MAX_MAGF);
    const float dm = pm - tm;
    d2 = fmaf(dm, dm, d2);
    spm += pm;
    stm += tm;
    spm2 = fmaf(pm, pm, spm2);
    stm2 = fmaf(tm, tm, stm2);

    // wrapped phase difference == angle(p' * conj(t')): one atan2, no mod.
    const float prb = pr + 1e-8f;
    const float trb = tr + 1e-8f;
    const float re = fmaf(prb, trb, pi * ti);
    const float im = fmaf(pi, trb, -(prb * ti));
    const float pd = atan2f(im, re);
    sph = fmaf(pd, pd, sph);

    // |p/(pm+1e-8) - t/(tm+1e-8)|^2
    const float rp = __builtin_amdgcn_rcpf(pm + 1e-8f);
    const float rt = __builtin_amdgcn_rcpf(tm + 1e-8f);
    const float dx = fmaf(pr, rp, -(tr * rt));
    const float dy = fmaf(pi, rp, -(ti * rt));
    scr = fmaf(dx, dx, fmaf(dy, dy, scr));
  }

  s_part[0][tid] = d2;
  s_part[1][tid] = spm;
  s_part[2][tid] = stm;
  s_part[3][tid] = spm2;
  s_part[4][tid] = stm2;
  s_part[5][tid] = sph;
  s_part[6][tid] = scr;
  s_part[7][tid] = sP;
  s_part[8][tid] = sQ;
  __syncthreads();

  if (tid < 32) {  // whole wave 0: EXEC all ones for WMMA
    const float P = wmma_reduce256(s_part[7]);
    const float Q = wmma_reduce256(s_part[8]);
    if (tid == 0) {
      s_PQ[0] = P;
      s_PQ[1] = Q;
    }
  }
  __syncthreads();

  const float invP = __builtin_amdgcn_rcpf(s_PQ[0]);
  const float invQ = __builtin_amdgcn_rcpf(s_PQ[1]);

  // JS divergence pass: entirely from LDS-resident magnitudes.
  float klp = 0.f, klq = 0.f;
#pragma unroll
  for (int w = 0; w < PER_THREAD; ++w) {
    const int s = tid + BLK * w;
    const float a = (s_magP[s] + EPSF) * invP;
    const float b = (s_magT[s] + EPSF) * invQ;
    const float r = __builtin_amdgcn_rcpf(a + b);
    // p*(log p - log m) = a * log(2a/(a+b)); likewise for q.
    klp = fmaf(a, __logf(2.0f * a * r), klp);
    klq = fmaf(b, __logf(2.0f * b * r), klq);
  }
  s_part[9][tid] = klp;
  s_part[10][tid] = klq;
  __syncthreads();

  if (tid < 32) {
    const float rN = 1.0f / (float)SUBC;
    const float S_d2 = wmma_reduce256(s_part[0]);
    const float S_pm = wmma_reduce256(s_part[1]);
    const float S_tm = wmma_reduce256(s_part[2]);
    const float S_pm2 = wmma_reduce256(s_part[3]);
    const float S_tm2 = wmma_reduce256(s_part[4]);
    const float S_ph = wmma_reduce256(s_part[5]);
    const float S_cr = wmma_reduce256(s_part[6]);
    const float S_klp = wmma_reduce256(s_part[9]);
    const float S_klq = wmma_reduce256(s_part[10]);
    if (tid == 0) {
      const float mean_p = S_pm * rN;
      const float mean_t = S_tm * rN;
      const float var_p = fmaf(-mean_p, mean_p, S_pm2 * rN);
      const float var_t = fmaf(-mean_t, mean_t, S_tm2 * rN);
      const float std_p = sqrtf(fmaxf(var_p, EPSF));
      const float std_t = sqrtf(fmaxf(var_t, EPSF));
      const float dmn = mean_p - mean_t;
      const float dsd = std_p - std_t;
      // weights: 0.5*magMSE + 0.25*meanDist + 0.15*stdDist + 0.5*phase
      //          + 0.2*corr + 0.1*JS  (JS = 0.5*(klp+klq))
      const float loss = 0.5f * (S_d2 * rN) + 0.25f * dmn * dmn +
                         0.15f * dsd * dsd + 0.5f * (S_ph * rN) +
                         0.2f * (S_cr * rN) + 0.05f * (S_klp + S_klq);
      row_loss[blockIdx.x] = loss;
    }
  }
}

__global__ __launch_bounds__(BLK) void csi_loss_final(
    const float* __restrict__ rows, float* __restrict__ out, int nrows) {
  __shared__ float s_part[BLK];
  float acc = 0.f;
  for (int i = threadIdx.x; i < nrows; i += BLK) acc += rows[i];
  s_part[threadIdx.x] = acc;
  __syncthreads();
  if (threadIdx.x < 32) {
    const float total = wmma_reduce256(s_part);
    if (threadIdx.x == 0) out[0] = total * (1.0f / (float)nrows);
  }
}

extern "C" void kernel_launch(void* const* d_in, const int* in_sizes, int n_in,
                              void* d_out, int out_size, void* d_ws,
                              size_t ws_size, hipStream_t stream) {
  const float* pr = (const float*)d_in[0];
  const float* pi = (const float*)d_in[1];
  const float* tr = (const float*)d_in[2];
  const float* ti = (const float*)d_in[3];
  const int total = in_sizes[0];
  const int nrows = total / SUBC;  // 32*64*8 = 16384 rows of 1024 subcarriers

  float* rows = (float*)d_ws;  // 16384 floats = 64 KB scratch, fully rewritten
  csi_loss_rows<<<nrows, BLK, 0, stream>>>(pr, pi, tr, ti, rows);
  csi_loss_final<<<1, BLK, 0, stream>>>(rows, (float*)d_out, nrows);
}